// LinearHopfieldCore_27075473834148
// MI455X (gfx1250) — compile-verified
//
#include <hip/hip_runtime.h>
#include <hip/hip_bf16.h>
#include <stdint.h>

typedef float v2f __attribute__((ext_vector_type(2)));
typedef float v8f __attribute__((ext_vector_type(8)));
typedef unsigned int u32x4 __attribute__((ext_vector_type(4)));
typedef int i32x4 __attribute__((ext_vector_type(4)));
typedef int i32x8 __attribute__((ext_vector_type(8)));

#define B_    4
#define L_    4096
#define S_    4096
#define H_    16
#define D_    64          // head dim == value dim E
#define EPSF  1e-6f
#define SPLIT 8           // S-dimension splits for phase-1 parallelism
#define SCHUNK (S_ / SPLIT)   // 512 s-values per block
#define PADR  72          // phase-1 LDS row stride (floats): 72%64==8 -> disjoint bank groups
#define PAD   68          // phase-2 LDS row stride (floats)

// ---------------------------------------------------------------------------
// Phase 1: partial KV^T = K^T (m.V) and Ksum = K^T m per (b,h,split).
// K chunk staged by the Tensor Data Mover (TDM); V staged by waves with the
// mask folded in. Grid: SPLIT*B*H blocks, 512 threads (16 waves, one 16x16
// KV tile each), fp32 WMMA.
// ---------------------------------------------------------------------------
__global__ __launch_bounds__(512)
void lh_phase1(const float* __restrict__ K, const float* __restrict__ V,
               const uint8_t* __restrict__ mask,
               float* __restrict__ kv_part,   // [SPLIT][B*H][64*64]  (layout e*64+d)
               float* __restrict__ ks_part)   // [SPLIT][B*H][64]
{
    __shared__ float KS[64 * PADR];    // [s][d] raw K chunk (TDM-written, padded rows)
    __shared__ float VS[64 * PADR];    // [s][d] masked V chunk
    __shared__ float KSP[512 * 4];     // per-thread 4-column Ksum partials

    const int t    = threadIdx.x;
    const int bh   = blockIdx.x % (B_ * H_);
    const int sp   = blockIdx.x / (B_ * H_);
    const int b    = bh / H_;
    const int h    = bh % H_;
    const int lane = t & 31;
    const int wave = t >> 5;              // 0..15
    const int d0   = (wave >> 2) * 16;    // output tile row (D)
    const int e0   = (wave & 3) * 16;     // output tile col (E)
    const int m    = lane & 15;
    const int kb   = (lane >> 4) * 2;     // K-dim sub-offset per half-wave

    const int dbase = (t & 15) * 4;       // 4 fixed d-columns owned by this thread
    const uint32_t ldsK = (uint32_t)(uintptr_t)&KS[0];   // LDS byte offset of KS

    // TDM descriptor group1 (chunk-invariant): 4-byte elems, pad 8 dwords every
    // 64 dwords (-> 72-float rows), 64x64 tile, dim0 stride = H*D.
    i32x8 g1;
    g1[0] = (2 << 16) | (1 << 20) | (5 << 22) | (7 << 25);
    g1[1] = (64 << 16);           // tensor_dim0[15:0] = 64
    g1[2] = (64 << 16);           // tensor_dim0 hi = 0, tensor_dim1[15:0] = 64
    g1[3] = (64 << 16);           // tensor_dim1 hi = 0, tile_dim0 = 64
    g1[4] = 64;                   // tile_dim1 = 64, tile_dim2 = 0
    g1[5] = H_ * D_;              // tensor_dim0_stride = 1024 elems
    g1[6] = 0;
    g1[7] = 0;
    const i32x4 gz4 = {0, 0, 0, 0};
    const i32x8 gz8 = {0, 0, 0, 0, 0, 0, 0, 0};

    v8f   acc = {};
    float ks4[4] = {0.f, 0.f, 0.f, 0.f};

    for (int c = 0; c < SCHUNK / 64; ++c) {          // 8 chunks of 64 s
        const int s0 = sp * SCHUNK + c * 64;
        __syncthreads();

        // --- kick the TDM for the raw K chunk (wave 0 only) ---
        if (wave == 0) {
            const uint64_t ga = (uint64_t)(uintptr_t)(K +
                ((size_t)(b * S_ + s0) * H_ + h) * D_);
            u32x4 g0;
            g0.x = 1u;                                     // count=1 (valid), load, no gather
            g0.y = ldsK;                                   // lds_addr
            g0.z = (uint32_t)ga;                           // global_addr[31:0]
            g0.w = (uint32_t)((ga >> 32) & 0x1FFFFFFu) | (2u << 30);  // addr[56:32] | type=2
            __builtin_amdgcn_tensor_load_to_lds(g0, g1, gz4, gz4, gz8, 0);
        }

        // --- stage masked V (all 512 threads) while the TDM streams K ---
        float mk[2];
        #pragma unroll
        for (int i = 0; i < 2; ++i) {
            const int q    = i * 512 + t;            // 0..1023 float4 slots
            const int srow = q >> 4;                 // 0..63
            const int s    = s0 + srow;
            const size_t g = ((size_t)(b * S_ + s) * H_ + h) * D_ + dbase;
            mk[i] = (float)mask[b * S_ + s];
            float4 vq = *(const float4*)(V + g);
            vq.x *= mk[i]; vq.y *= mk[i]; vq.z *= mk[i]; vq.w *= mk[i];
            *(float4*)&VS[srow * PADR + dbase] = vq;
        }

        if (wave == 0) __builtin_amdgcn_s_wait_tensorcnt(0);
        __syncthreads();

        // --- Ksum: masked K read back from the TDM-filled LDS tile ---
        #pragma unroll
        for (int i = 0; i < 2; ++i) {
            const int srow = (i * 512 + t) >> 4;
            const float4 kq = *(const float4*)&KS[srow * PADR + dbase];
            ks4[0] += kq.x * mk[i]; ks4[1] += kq.y * mk[i];
            ks4[2] += kq.z * mk[i]; ks4[3] += kq.w * mk[i];
        }

        // --- 16 WMMA steps of K=4 (KV = K^T (mV): mask carried by V) ---
        #pragma unroll
        for (int st = 0; st < 16; ++st) {
            const int sc = st * 4 + kb;
            v2f a, bb;
            a.x  = KS[(sc + 0) * PADR + d0 + m];
            a.y  = KS[(sc + 1) * PADR + d0 + m];
            bb.x = VS[(sc + 0) * PADR + e0 + m];
            bb.y = VS[(sc + 1) * PADR + e0 + m];
            acc = __builtin_amdgcn_wmma_f32_16x16x4_f32(
                      false, a, false, bb, (short)0, acc, false, false);
        }
    }

    // store partial KV^T tile: element (d = d0+j(+8), e = e0+m)
    float* kvp = kv_part + (size_t)(sp * (B_ * H_) + bh) * (64 * 64);
    #pragma unroll
    for (int j = 0; j < 8; ++j) {
        const int dd = d0 + j + ((lane >> 4) << 3);
        kvp[(e0 + m) * 64 + dd] = acc[j];
    }

    // deterministic fixed-order Ksum reduction.
    // column d is owned by threads with (t&15) == d>>2, component d&3.
    #pragma unroll
    for (int j = 0; j < 4; ++j) KSP[t * 4 + j] = ks4[j];
    __syncthreads();
    if (t < 64) {
        const int tg   = t >> 2;
        const int comp = t & 3;
        float s = 0.f;
        #pragma unroll
        for (int w = 0; w < 32; ++w) s += KSP[(w * 16 + tg) * 4 + comp];
        ks_part[(size_t)(sp * (B_ * H_) + bh) * 64 + t] = s;
    }
}

// ---------------------------------------------------------------------------
// Reduce the SPLIT partials (fixed order -> deterministic).
// ---------------------------------------------------------------------------
__global__ __launch_bounds__(256)
void lh_reduce(const float* __restrict__ kv_part, const float* __restrict__ ks_part,
               float* __restrict__ kv, float* __restrict__ ks)
{
    const int idx = blockIdx.x * 256 + threadIdx.x;   // float4 index
    if (idx < (64 * 4096) / 4) {
        float4 s = {0.f, 0.f, 0.f, 0.f};
        #pragma unroll
        for (int i = 0; i < SPLIT; ++i) {
            const float4 p = ((const float4*)kv_part)[(size_t)i * (64 * 1024) + idx];
            s.x += p.x; s.y += p.y; s.z += p.z; s.w += p.w;
        }
        ((float4*)kv)[idx] = s;
    }
    if (idx < 64 * 64) {
        float s = 0.f;
        #pragma unroll
        for (int i = 0; i < SPLIT; ++i) s += ks_part[(size_t)i * (64 * 64) + idx];
        ks[idx] = s;
    }
}

// ---------------------------------------------------------------------------
// Phase 2: out = (Q @ KV) / (Q . (Ksum+eps)), fp32 WMMA.
// Grid: B*H*32 blocks (128 L-rows each), 256 threads (8 waves x 16 rows).
// ---------------------------------------------------------------------------
__global__ __launch_bounds__(256)
void lh_phase2(const float* __restrict__ Q, const float* __restrict__ kv,
               const float* __restrict__ ks, float* __restrict__ out)
{
    __shared__ float Qs[128 * PAD];    // [row][d]
    __shared__ float KVTs[64 * PAD];   // [e][d]
    __shared__ float KsumS[64];
    __shared__ float RDen[128];

    const int t    = threadIdx.x;
    const int bh   = blockIdx.x >> 5;
    const int lc   = blockIdx.x & 31;
    const int b    = bh / H_;
    const int h    = bh % H_;
    const int l0   = lc * 128;
    const int lane = t & 31;
    const int wave = t >> 5;              // 0..7
    const int m0   = wave * 16;
    const int m    = lane & 15;
    const int kb   = (lane >> 4) * 2;

    // stage KV^T (1024 float4, coalesced b128 both sides)
    #pragma unroll
    for (int i = 0; i < 4; ++i) {
        const int q = i * 256 + t;
        const int e = q >> 4;
        const int d = (q & 15) * 4;
        *(float4*)&KVTs[e * PAD + d] =
            *(const float4*)&kv[(size_t)bh * 4096 + q * 4];
    }
    if (t < 64) KsumS[t] = ks[(size_t)bh * 64 + t] + EPSF;
    // stage 128 Q rows (2048 float4)
    #pragma unroll
    for (int i = 0; i < 8; ++i) {
        const int q = i * 256 + t;
        const int r = q >> 4;
        const int d = (q & 15) * 4;
        *(float4*)&Qs[r * PAD + d] =
            *(const float4*)&Q[((size_t)(b * L_ + l0 + r) * H_ + h) * D_ + d];
    }
    __syncthreads();

    // denominators: lanes 0..15 of each wave -> rows m0+lane (stride 68 -> conflict-free)
    if (lane < 16) {
        const float* qr = &Qs[(m0 + lane) * PAD];
        float dsum = 0.f;
        #pragma unroll
        for (int d = 0; d < 64; ++d) dsum += qr[d] * KsumS[d];
        RDen[m0 + lane] = 1.0f / dsum;
    }
    __syncthreads();

    const int mbase = m0 + ((lane >> 4) << 3);
    float rd[8];
    #pragma unroll
    for (int j = 0; j < 8; ++j) rd[j] = RDen[mbase + j];

    v8f acc0 = {}, acc1 = {}, acc2 = {}, acc3 = {};
    #pragma unroll
    for (int kk = 0; kk < 16; ++kk) {                // K-dim (D) in steps of 4
        const int c = kk * 4 + kb;
        v2f a  = *(const v2f*)&Qs[(m0 + m) * PAD + c];
        v2f b0 = *(const v2f*)&KVTs[( 0 + m) * PAD + c];
        v2f b1 = *(const v2f*)&KVTs[(16 + m) * PAD + c];
        v2f b2 = *(const v2f*)&KVTs[(32 + m) * PAD + c];
        v2f b3 = *(const v2f*)&KVTs[(48 + m) * PAD + c];
        acc0 = __builtin_amdgcn_wmma_f32_16x16x4_f32(false, a, false, b0, (short)0, acc0, false, false);
        acc1 = __builtin_amdgcn_wmma_f32_16x16x4_f32(false, a, false, b1, (short)0, acc1, false, false);
        acc2 = __builtin_amdgcn_wmma_f32_16x16x4_f32(false, a, false, b2, (short)0, acc2, false, false);
        acc3 = __builtin_amdgcn_wmma_f32_16x16x4_f32(false, a, false, b3, (short)0, acc3, false, false);
    }

    // scale by 1/denom and store: acc[j] -> (row mbase+j, e = et*16 + m)
    #pragma unroll
    for (int j = 0; j < 8; ++j) {
        float* orow = out + ((size_t)(b * L_ + l0 + mbase + j) * H_ + h) * D_;
        orow[ 0 + m] = acc0[j] * rd[j];
        orow[16 + m] = acc1[j] * rd[j];
        orow[32 + m] = acc2[j] * rd[j];
        orow[48 + m] = acc3[j] * rd[j];
    }
}

// ---------------------------------------------------------------------------
extern "C" void kernel_launch(void* const* d_in, const int* in_sizes, int n_in,
                              void* d_out, int out_size, void* d_ws, size_t ws_size,
                              hipStream_t stream) {
    const float*   Q    = (const float*)d_in[0];
    const float*   K    = (const float*)d_in[1];
    const float*   V    = (const float*)d_in[2];
    const uint8_t* mask = (const uint8_t*)d_in[3];
    float* out = (float*)d_out;

    float* kv_part = (float*)d_ws;                                   // SPLIT*64*4096
    float* ks_part = kv_part + (size_t)SPLIT * 64 * 4096;            // SPLIT*64*64
    float* kvfin   = ks_part + (size_t)SPLIT * 64 * 64;              // 64*4096
    float* ksfin   = kvfin   + (size_t)64 * 4096;                    // 64*64

    lh_phase1<<<SPLIT * B_ * H_, 512, 0, stream>>>(K, V, mask, kv_part, ks_part);
    lh_reduce<<<(64 * 4096 / 4) / 256, 256, 0, stream>>>(kv_part, ks_part, kvfin, ksfin);
    lh_phase2<<<B_ * H_ * (L_ / 128), 256, 0, stream>>>(Q, kvfin, ksfin, out);
}